// SepConv2dComb_37306085933387
// MI455X (gfx1250) — compile-verified
//
#include <hip/hip_runtime.h>

#define B_DIM  8
#define CIN    32
#define COUT   64
#define H_DIM  64
#define W_DIM  64
#define HW     (H_DIM * W_DIM)        // 4096
#define NTOT   (B_DIM * COUT * HW)    // 2,097,152 output elems
#define EPS_BN 1e-5f

typedef __attribute__((ext_vector_type(2))) float v2f;
typedef __attribute__((ext_vector_type(8))) float v8f;

// ---------------------------------------------------------------------------
// Kernel 0: zero the per-channel stats accumulators (sum, sumsq) in workspace
// ---------------------------------------------------------------------------
__global__ void zero_stats_kernel(float* __restrict__ stats) {
    if (threadIdx.x < 2 * COUT) stats[threadIdx.x] = 0.0f;
}

// ---------------------------------------------------------------------------
// Kernel 1: u[b,j,n] = sum_i W2[j,i] * x[b,i,n]   (1x1 channel mix as GEMM)
// One wave32 per 16x16 output tile; K=32 done as 8 chained
// V_WMMA_F32_16X16X4_F32 ops (exact f32 math — problem is HBM-bound, no
// reason to drop precision).
//
// VGPR layouts per CDNA5 ISA 7.12.2 (f32, 16x16x4):
//   A (16x4, MxK): lanes 0-15 -> M=lane, v0=K0 v1=K1 ; lanes 16-31 -> v0=K2 v1=K3
//   B (4x16, KxN): lanes 0-15 -> N=lane, v0=K0 v1=K1 ; lanes 16-31 -> v0=K2 v1=K3
//   C/D (16x16):   vgpr r: lanes 0-15 -> M=r ; lanes 16-31 -> M=r+8 ; N=lane&15
// ---------------------------------------------------------------------------
__global__ __launch_bounds__(256) void gemm_wmma_kernel(
    const float* __restrict__ x,     // [B, CIN, HW]
    const float* __restrict__ W2,    // [COUT, CIN]
    float* __restrict__ u)           // [B, COUT, HW]
{
    const int lane = threadIdx.x & 31;
    const int wave = threadIdx.x >> 5;
    const int tile = blockIdx.x * 8 + wave;      // 8192 tiles total

    const int b   = tile >> 10;                  // (COUT/16)*(HW/16) = 1024 tiles/batch
    const int r   = tile & 1023;
    const int m0  = (r >> 8) << 4;               // Cout tile base (0,16,32,48)
    const int n0  = (r & 255) << 4;              // HW tile base

    const int half = lane >> 4;                  // 0: lanes 0-15, 1: lanes 16-31
    const int l15  = lane & 15;

    const float* xb = x + (size_t)b * CIN * HW;

    v8f c = {};
    #pragma unroll
    for (int kk = 0; kk < CIN; kk += 4) {
        const int ka = kk + half * 2;            // this half-wave's K columns
        v2f a, bb;
        a.x  = W2[(m0 + l15) * CIN + ka];
        a.y  = W2[(m0 + l15) * CIN + ka + 1];
        bb.x = xb[(size_t)ka       * HW + n0 + l15];
        bb.y = xb[(size_t)(ka + 1) * HW + n0 + l15];
        // (neg_a, A, neg_b, B, c_mod, C, reuse_a, reuse_b)
        c = __builtin_amdgcn_wmma_f32_16x16x4_f32(false, a, false, bb,
                                                  (short)0, c, false, false);
    }

    float* ub = u + (size_t)b * COUT * HW;
    const int row0 = m0 + half * 8;
    #pragma unroll
    for (int rr = 0; rr < 8; ++rr) {
        ub[(size_t)(row0 + rr) * HW + n0 + l15] = c[rr];
    }
}

// ---------------------------------------------------------------------------
// Kernel 2: depthwise 3x3 (pad 1) + bias -> z (written to d_out pre-BN),
// plus per-channel sum / sumsq accumulation for training-mode batchnorm.
// Each block covers 256 contiguous pixels of one (b, j) plane, so the block
// reduction yields a single channel's partial sums -> 2 float atomics/block.
// ---------------------------------------------------------------------------
__global__ __launch_bounds__(256) void dwconv_stats_kernel(
    const float* __restrict__ u,     // [B, COUT, HW]
    const float* __restrict__ W1,    // [COUT, 1, 3, 3]
    const float* __restrict__ b2,    // [COUT]
    float* __restrict__ z,           // [B, COUT, HW]  (= d_out, pre-BN)
    float* __restrict__ stats)       // [2*COUT]
{
    __shared__ float s1[256];
    __shared__ float s2[256];

    const int plane = blockIdx.x >> 4;           // (b, j) plane index
    const int chunk = blockIdx.x & 15;
    const int j     = plane & (COUT - 1);
    const int n     = chunk * 256 + threadIdx.x;
    const int h     = n >> 6;
    const int w     = n & 63;

    const float* up = u + (size_t)plane * HW;
    const float* k  = W1 + j * 9;

    float acc = b2[j];
    #pragma unroll
    for (int dy = -1; dy <= 1; ++dy) {
        const int hh = h + dy;
        if (hh < 0 || hh >= H_DIM) continue;
        #pragma unroll
        for (int dx = -1; dx <= 1; ++dx) {
            const int ww = w + dx;
            if (ww < 0 || ww >= W_DIM) continue;
            acc += up[hh * W_DIM + ww] * k[(dy + 1) * 3 + (dx + 1)];
        }
    }

    z[(size_t)plane * HW + n] = acc;

    s1[threadIdx.x] = acc;
    s2[threadIdx.x] = acc * acc;
    __syncthreads();
    #pragma unroll
    for (int s = 128; s > 0; s >>= 1) {
        if (threadIdx.x < s) {
            s1[threadIdx.x] += s1[threadIdx.x + s];
            s2[threadIdx.x] += s2[threadIdx.x + s];
        }
        __syncthreads();
    }
    if (threadIdx.x == 0) {
        atomicAdd(&stats[j],        s1[0]);
        atomicAdd(&stats[COUT + j], s2[0]);
    }
}

// ---------------------------------------------------------------------------
// Kernel 3: batchnorm (batch statistics) + affine + ReLU, in place on d_out.
// ---------------------------------------------------------------------------
__global__ __launch_bounds__(256) void bn_relu_kernel(
    const float* __restrict__ stats,
    const float* __restrict__ gamma,
    const float* __restrict__ beta,
    float* __restrict__ out)
{
    const int idx = blockIdx.x * 256 + threadIdx.x;
    if (idx >= NTOT) return;
    const int j = (idx >> 12) & (COUT - 1);      // layout: b*COUT*HW + j*HW + n

    const float invN = 1.0f / (float)(B_DIM * HW);
    const float mean = stats[j] * invN;
    const float var  = stats[COUT + j] * invN - mean * mean;
    const float sc   = rsqrtf(var + EPS_BN) * gamma[j];
    const float v    = (out[idx] - mean) * sc + beta[j];
    out[idx] = v > 0.0f ? v : 0.0f;
}

// ---------------------------------------------------------------------------
// Host-side launcher. Inputs (setup_inputs order): x, W1, W2, b2, gamma, beta
// Workspace: u  at float offset 0            (B*COUT*HW floats = 8 MB)
//            stats at float offset B*COUT*HW (2*COUT floats)
// ---------------------------------------------------------------------------
extern "C" void kernel_launch(void* const* d_in, const int* in_sizes, int n_in,
                              void* d_out, int out_size, void* d_ws, size_t ws_size,
                              hipStream_t stream) {
    (void)in_sizes; (void)n_in; (void)out_size; (void)ws_size;

    const float* x     = (const float*)d_in[0];
    const float* W1    = (const float*)d_in[1];
    const float* W2    = (const float*)d_in[2];
    const float* b2    = (const float*)d_in[3];
    const float* gamma = (const float*)d_in[4];
    const float* beta  = (const float*)d_in[5];

    float* u     = (float*)d_ws;
    float* stats = u + (size_t)B_DIM * COUT * HW;
    float* out   = (float*)d_out;

    zero_stats_kernel<<<1, 128, 0, stream>>>(stats);

    // 8 * (COUT/16) * (HW/16) = 8192 wave-tiles, 8 waves per 256-thread block
    gemm_wmma_kernel<<<1024, 256, 0, stream>>>(x, W2, u);

    // one block per 256 pixels of each (b, j) plane: 8*64*16 = 8192 blocks
    dwconv_stats_kernel<<<8192, 256, 0, stream>>>(u, W1, b2, out, stats);

    bn_relu_kernel<<<(NTOT + 255) / 256, 256, 0, stream>>>(stats, gamma, beta, out);
}